// DBPGCN_41059887350098
// MI455X (gfx1250) — compile-verified
//
#include <hip/hip_runtime.h>
#include <hip/hip_bf16.h>

// ---------------------------------------------------------------------------
// DBP-GCN on MI455X (gfx1250, wave32).
//  - GEMMs via V_WMMA_F32_16X16X4_F32; one wave owns a 16 x FOUT output strip
//    so the A fragment is loaded once per k-step and reused across all column
//    tiles (8x less A traffic than tile-per-wave).
//  - Edge aggregation is the hot loop: 800k random 512B/256B row gathers
//    (b128 loads) + f32 atomic scatter-adds. h1 (25.6MB) / h2 (12.8MB) fit in
//    the 192MB L2, so this is L2-bound.
// ---------------------------------------------------------------------------

typedef float v2f __attribute__((ext_vector_type(2)));
typedef float v8f __attribute__((ext_vector_type(8)));

// ---------------- utility: zero a float range (grid-stride) ----------------
__global__ void dbp_zero_kernel(float* __restrict__ p, long n) {
    long i = (long)blockIdx.x * blockDim.x + threadIdx.x;
    long stride = (long)gridDim.x * blockDim.x;
    for (; i < n; i += stride) p[i] = 0.0f;
}

// ---------------- degree + inverse-sqrt norm -------------------------------
__global__ void dbp_degree_kernel(const int* __restrict__ dst, float* __restrict__ deg, int E) {
    int i = blockIdx.x * blockDim.x + threadIdx.x;
    if (i < E) atomicAdd(&deg[dst[i]], 1.0f);
}

__global__ void dbp_dinv_kernel(float* __restrict__ deg, int N) {
    int i = blockIdx.x * blockDim.x + threadIdx.x;
    if (i < N) deg[i] = rsqrtf(deg[i] + 1.0f);   // in place: deg -> dinv
}

// ---------------- fp32 WMMA GEMM: C[M,FOUT] = op(A[M,128]) * B[128,FOUT] ----
// One wave owns a 16 x FOUT strip: TILES = FOUT/16 accumulators, A fragment
// shared across column tiles each k-step.  Fragment layout per CDNA5 ISA
// (16x4 f32 A): VGPR v, lane-half hi -> K = v + 2*hi, M = lane%16, so the two
// per-lane A values are consecutive -> one b64 load.
template <int FOUT, bool RELU_IN>
__global__ __launch_bounds__(256) void dbp_wmma_gemm_kernel(
    const float* __restrict__ A, const float* __restrict__ B,
    float* __restrict__ C, int M) {
    constexpr int TILES = FOUT / 16;
    const int wave = (blockIdx.x * blockDim.x + threadIdx.x) >> 5;
    const int lane = threadIdx.x & 31;
    if (wave >= M / 16) return;                  // uniform per wave: EXEC all-1s

    const int l16 = lane & 15;
    const int hi  = lane >> 4;                   // 0 or 1
    const float* arow = A + (size_t)(wave * 16 + l16) * 128;

    v8f acc[TILES];
    v8f zv = {};
#pragma unroll
    for (int t = 0; t < TILES; ++t) acc[t] = zv;

    for (int kb = 0; kb < 128; kb += 4) {
        const int k0 = kb + 2 * hi;              // K = v + 2*hi, v in {0,1}
        v2f a = *(const v2f*)(arow + k0);        // 8B aligned (k0 even)
        if (RELU_IN) { a.x = fmaxf(a.x, 0.0f); a.y = fmaxf(a.y, 0.0f); }
        const float* brow0 = B + (size_t)(k0 + 0) * FOUT + l16;
        const float* brow1 = B + (size_t)(k0 + 1) * FOUT + l16;
#pragma unroll
        for (int t = 0; t < TILES; ++t) {
            v2f b;
            b.x = brow0[t * 16];
            b.y = brow1[t * 16];
            acc[t] = __builtin_amdgcn_wmma_f32_16x16x4_f32(
                /*neg_a=*/false, a, /*neg_b=*/false, b,
                /*c_mod=*/(short)0, acc[t], /*reuse_a=*/false, /*reuse_b=*/false);
        }
    }
    // C/D layout: VGPR r -> row wave*16 + r + 8*hi, col t*16 + l16
#pragma unroll
    for (int t = 0; t < TILES; ++t) {
        float* ccol = C + (size_t)(wave * 16 + 8 * hi) * FOUT + t * 16 + l16;
#pragma unroll
        for (int r = 0; r < 8; ++r)
            ccol[(size_t)r * FOUT] = acc[t][r];
    }
}

// ---------------- edge scatter-add: agg[dst] += h[src] * norm --------------
// One wave per edge; one b128/b64 row-gather per lane, per-dword f32 atomics.
template <int F>
__global__ __launch_bounds__(256) void dbp_edge_agg_kernel(
    const int* __restrict__ src, const int* __restrict__ dst,
    const float* __restrict__ dinv, const float* __restrict__ h,
    float* __restrict__ agg, int E) {
    const int gid  = blockIdx.x * blockDim.x + threadIdx.x;
    const int edge = gid >> 5;
    const int lane = gid & 31;
    if (edge >= E) return;
    const int s = src[edge];
    const int d = dst[edge];
    const float norm = dinv[s] * dinv[d];
    const float* hrow = h   + (size_t)s * F;
    float*       arow = agg + (size_t)d * F;
    if constexpr (F == 128) {
        const float4 hv = *(const float4*)(hrow + lane * 4);   // 16B aligned
        atomicAdd(&arow[lane * 4 + 0], hv.x * norm);
        atomicAdd(&arow[lane * 4 + 1], hv.y * norm);
        atomicAdd(&arow[lane * 4 + 2], hv.z * norm);
        atomicAdd(&arow[lane * 4 + 3], hv.w * norm);
    } else {
        const float2 hv = *(const float2*)(hrow + lane * 2);   // 8B aligned
        atomicAdd(&arow[lane * 2 + 0], hv.x * norm);
        atomicAdd(&arow[lane * 2 + 1], hv.y * norm);
    }
}

// ---------------- self-loop + bias + relu for layer 1, in place ------------
__global__ void dbp_selfloop_bias_relu_kernel(
    float* __restrict__ agg, const float* __restrict__ h,
    const float* __restrict__ dinv, const float* __restrict__ bias,
    int M, int F) {
    long i = (long)blockIdx.x * blockDim.x + threadIdx.x;
    if (i >= (long)M * F) return;
    const int node = (int)(i / F);
    const int f    = (int)(i % F);
    const float di = dinv[node];
    float v = agg[i] + h[i] * (di * di) + bias[f];
    agg[i] = fmaxf(v, 0.0f);
}

// ---------------- layer 2 epilogue: self-loop + bias + softmax(64) ---------
// One wave per node row; 2 elements/lane; shfl_xor wave reductions.
__global__ __launch_bounds__(256) void dbp_softmax_kernel(
    const float* __restrict__ agg, const float* __restrict__ h,
    const float* __restrict__ dinv, const float* __restrict__ bias,
    float* __restrict__ out, int M) {
    const int gid  = blockIdx.x * blockDim.x + threadIdx.x;
    const int row  = gid >> 5;
    const int lane = gid & 31;
    if (row >= M) return;
    const float di = dinv[row];
    const float dd = di * di;
    const float* a  = agg + (size_t)row * 64;
    const float* hr = h   + (size_t)row * 64;
    float v0 = a[lane]      + hr[lane]      * dd + bias[lane];
    float v1 = a[lane + 32] + hr[lane + 32] * dd + bias[lane + 32];
    float mx = fmaxf(v0, v1);
#pragma unroll
    for (int off = 16; off > 0; off >>= 1) mx = fmaxf(mx, __shfl_xor(mx, off, 32));
    const float e0 = __expf(v0 - mx);
    const float e1 = __expf(v1 - mx);
    float s = e0 + e1;
#pragma unroll
    for (int off = 16; off > 0; off >>= 1) s += __shfl_xor(s, off, 32);
    const float inv = 1.0f / s;
    float* o = out + (size_t)row * 64;
    o[lane]      = e0 * inv;
    o[lane + 32] = e1 * inv;
}

// ---------------------------------------------------------------------------
extern "C" void kernel_launch(void* const* d_in, const int* in_sizes, int n_in,
                              void* d_out, int out_size, void* d_ws, size_t ws_size,
                              hipStream_t stream) {
    const float* x  = (const float*)d_in[0];   // [N,128]
    const int*   ei = (const int*)d_in[1];     // [2,E]
    const float* W1 = (const float*)d_in[2];   // [128,128]
    const float* b1 = (const float*)d_in[3];   // [128]
    const float* W2 = (const float*)d_in[4];   // [128,64]
    const float* b2 = (const float*)d_in[5];   // [64]
    float* out = (float*)d_out;                // [N,64]

    const int N = in_sizes[0] / 128;
    const int E = in_sizes[1] / 2;
    const int* srcIdx = ei;                    // edge_index[0]
    const int* dstIdx = ei + E;                // edge_index[1]

    // workspace layout (floats): dinv | h1 | agg1(->x1) | h2 | agg2
    float* ws   = (float*)d_ws;
    float* dinv = ws;
    float* h1   = dinv + (((size_t)N + 255) & ~(size_t)255);
    float* agg1 = h1   + (size_t)N * 128;
    float* h2   = agg1 + (size_t)N * 128;
    float* agg2 = h2   + (size_t)N * 64;

    const int TB = 256;
    auto blocksFor = [](long n, int tb) { return (int)((n + tb - 1) / tb); };

    // 1. zero accumulators (atomics below must start from zero every call)
    dbp_zero_kernel<<<blocksFor(N, TB), TB, 0, stream>>>(dinv, N);
    dbp_zero_kernel<<<blocksFor((long)N * 128, TB), TB, 0, stream>>>(agg1, (long)N * 128);
    dbp_zero_kernel<<<blocksFor((long)N * 64, TB), TB, 0, stream>>>(agg2, (long)N * 64);

    // 2. degree -> dinv
    dbp_degree_kernel<<<blocksFor(E, TB), TB, 0, stream>>>(dstIdx, dinv, E);
    dbp_dinv_kernel<<<blocksFor(N, TB), TB, 0, stream>>>(dinv, N);

    // 3. h1 = relu(x) @ W1   (WMMA, one wave per 16x128 strip)
    dbp_wmma_gemm_kernel<128, true>
        <<<blocksFor((long)(N / 16) * 32, TB), TB, 0, stream>>>(x, W1, h1, N);

    // 4. agg1[dst] += h1[src] * norm
    dbp_edge_agg_kernel<128>
        <<<blocksFor((long)E * 32, TB), TB, 0, stream>>>(srcIdx, dstIdx, dinv, h1, agg1, E);

    // 5. x1 = relu(agg1 + h1*dinv^2 + b1)   (in place in agg1)
    dbp_selfloop_bias_relu_kernel
        <<<blocksFor((long)N * 128, TB), TB, 0, stream>>>(agg1, h1, dinv, b1, N, 128);

    // 6. h2 = x1 @ W2   (one wave per 16x64 strip)
    dbp_wmma_gemm_kernel<64, false>
        <<<blocksFor((long)(N / 16) * 32, TB), TB, 0, stream>>>(agg1, W2, h2, N);

    // 7. agg2[dst] += h2[src] * norm
    dbp_edge_agg_kernel<64>
        <<<blocksFor((long)E * 32, TB), TB, 0, stream>>>(srcIdx, dstIdx, dinv, h2, agg2, E);

    // 8. out = softmax(agg2 + h2*dinv^2 + b2)
    dbp_softmax_kernel
        <<<blocksFor((long)N * 32, TB), TB, 0, stream>>>(agg2, h2, dinv, b2, out, N);
}